// TimmViTEncoder_89326729822471
// MI455X (gfx1250) — compile-verified
//
#include <hip/hip_runtime.h>

// ---------------------------------------------------------------------------
// Problem constants (from reference)
// ---------------------------------------------------------------------------
#define NB     256   // batch
#define CIN    3
#define IMGS   128
#define PS     8
#define NIMG   256   // image tokens (16x16 patches)
#define LTOK   16    // latent tokens
#define CH     48    // channels
#define NH     4     // heads
#define HDIM   12    // head dim
#define NTOK   272   // NIMG + LTOK
#define PRUNE_D 6
#define NMASK  128

typedef float v2f __attribute__((ext_vector_type(2)));
typedef float v8f __attribute__((ext_vector_type(8)));

// Async global->LDS path (CDNA5 GLOBAL_LOAD_ASYNC_TO_LDS_B32), guarded so the
// file compiles even if this toolchain does not declare the builtins.
#ifndef __has_builtin
#define __has_builtin(x) 0
#endif
#if __has_builtin(__builtin_amdgcn_global_load_async_to_lds_b32) && \
    __has_builtin(__builtin_amdgcn_s_wait_asynccnt)
#define HAVE_ASYNC_LDS 1
#else
#define HAVE_ASYNC_LDS 0
#endif

#if HAVE_ASYNC_LDS
typedef __attribute__((address_space(1))) int g_as_int;
typedef __attribute__((address_space(3))) int l_as_int;
#endif

__device__ __forceinline__ void async_g2l_b32(const float* g, float* l) {
#if HAVE_ASYNC_LDS
  __builtin_amdgcn_global_load_async_to_lds_b32((g_as_int*)g, (l_as_int*)l, 0,
                                                0);
#else
  *l = *g;
#endif
}
__device__ __forceinline__ void async_wait_all() {
#if HAVE_ASYNC_LDS
  __builtin_amdgcn_s_wait_asynccnt(0);
#endif
}

// D(16x16,f32) += A(16x4,f32) * B(4x16,f32)   -> v_wmma_f32_16x16x4_f32
__device__ __forceinline__ v8f wmma4(v2f a, v2f b, v8f c) {
  return __builtin_amdgcn_wmma_f32_16x16x4_f32(false, a, false, b, (short)0, c,
                                               false, false);
}
__device__ __forceinline__ v8f vzero8() {
  v8f z;
  for (int i = 0; i < 8; ++i) z[i] = 0.f;
  return z;
}
__device__ __forceinline__ float geluf(float x) {
  return 0.5f * x * (1.f + erff(x * 0.70710678118654752f));
}

// ---------------------------------------------------------------------------
// 1) mask flags: maskflag[b][n] = 1 if n in orders[b][:NMASK]
// ---------------------------------------------------------------------------
__global__ __launch_bounds__(256) void k_mask(const int* __restrict__ orders,
                                              float* __restrict__ maskflag) {
  int b = blockIdx.x, t = threadIdx.x;
  maskflag[b * NIMG + t] = 0.f;
  __syncthreads();
  if (t < NMASK) maskflag[b * NIMG + orders[b * NIMG + t]] = 1.f;
}

// ---------------------------------------------------------------------------
// 2) patch-embed conv (as 256x48 GEMM, K=192) + mask/pos/latent + LN -> xs
// ---------------------------------------------------------------------------
__global__ __launch_bounds__(256) void k_embed(
    const float* __restrict__ x, const float* __restrict__ conv_w,
    const float* __restrict__ conv_b, const float* __restrict__ pos_embed,
    const float* __restrict__ mask_token, const float* __restrict__ lat_tok,
    const float* __restrict__ lat_pos, const float* __restrict__ norm_g,
    const float* __restrict__ norm_b, const float* __restrict__ maskflag,
    float* __restrict__ xs) {
  __shared__ float conv_lds[NIMG][CH];
  int b = blockIdx.x, tid = threadIdx.x;
  int lane = tid & 31, wave = tid >> 5;
  int fr = lane & 15;
  int koff = (lane & 16) ? 2 : 0;
  int rbase = (lane & 16) ? 8 : 0;
  const float* xb = x + (size_t)b * CIN * IMGS * IMGS;

  // GEMM: A[patch][k] (k = ci*64+py*8+px), B[k][c] = conv_w[c*192+k]
  for (int tile = wave; tile < 48; tile += 8) {  // 16 M-tiles x 3 N-tiles
    int tm = tile / 3, tn = tile % 3;
    int m = tm * 16 + fr;
    int gy = m >> 4, gx = m & 15;
    const float* arow = xb + (gy * PS) * IMGS + gx * PS;
    int n = tn * 16 + fr;
    const float* brow = conv_w + n * 192;
    v8f acc = vzero8();
    for (int k = 0; k < 192; k += 4) {
      int kk = k + koff;
      int ci = kk >> 6, rem = kk & 63, py = rem >> 3, px = rem & 7;
      const float* ap = arow + ci * (IMGS * IMGS) + py * IMGS + px;
      v2f a;  a.x = ap[0];       a.y = ap[1];        // contiguous in px
      v2f bb; bb.x = brow[kk];   bb.y = brow[kk + 1];
      acc = wmma4(a, bb, acc);
    }
    for (int r = 0; r < 8; ++r)
      conv_lds[tm * 16 + rbase + r][n] = acc[r] + conv_b[n];
  }
  __syncthreads();

  // assemble token rows + LayerNorm(norm_g, norm_b)
  for (int nrow = tid; nrow < NTOK; nrow += 256) {
    float row[CH];
    if (nrow < NIMG) {
      float mf = maskflag[b * NIMG + nrow];
      for (int c = 0; c < CH; ++c) {
        float t = (mf > 0.f) ? mask_token[c] : conv_lds[nrow][c];
        row[c] = t + pos_embed[nrow * CH + c];
      }
    } else {
      int l = nrow - NIMG;
      for (int c = 0; c < CH; ++c)
        row[c] = lat_tok[l * CH + c] + lat_pos[l * CH + c];
    }
    float mu = 0.f;
    for (int c = 0; c < CH; ++c) mu += row[c];
    mu *= (1.f / CH);
    float var = 0.f;
    for (int c = 0; c < CH; ++c) { float d = row[c] - mu; var += d * d; }
    var *= (1.f / CH);
    float rstd = rsqrtf(var + 1e-5f);
    float* dst = xs + ((size_t)b * NTOK + nrow) * CH;
    for (int c = 0; c < CH; ++c)
      dst[c] = (row[c] - mu) * rstd * norm_g[c] + norm_b[c];
  }
}

// ---------------------------------------------------------------------------
// 3) score path -> binary keep policy (forward value of straight-through)
//    Deterministic column-mean reduction (no float atomics).
// ---------------------------------------------------------------------------
__global__ __launch_bounds__(256) void k_score(
    const float* __restrict__ xs, const float* __restrict__ g_ln,
    const float* __restrict__ b_ln, const float* __restrict__ w_in,
    const float* __restrict__ b_in, const float* __restrict__ w1,
    const float* __restrict__ b1, const float* __restrict__ w2,
    const float* __restrict__ b2, const float* __restrict__ w3,
    const float* __restrict__ b3, const float* __restrict__ gumbel_u,
    float* __restrict__ policy) {
  __shared__ float hn[NTOK][CH];
  __shared__ float colpart[32][24];  // [tileM*2 + half][col-24] partial sums
  __shared__ float colsum[24];
  __shared__ float feat[LTOK][CH];
  __shared__ float logits[LTOK];
  __shared__ int lpos;
  int b = blockIdx.x, tid = threadIdx.x;
  int lane = tid & 31, wave = tid >> 5;
  int fr = lane & 15;
  int koff = (lane & 16) ? 2 : 0;
  int rbase = (lane & 16) ? 8 : 0;
  int half = (lane & 16) ? 1 : 0;

  for (int nrow = tid; nrow < NTOK; nrow += 256) {   // sp_ln LayerNorm
    const float* src = xs + ((size_t)b * NTOK + nrow) * CH;
    float mu = 0.f;
    for (int c = 0; c < CH; ++c) mu += src[c];
    mu *= (1.f / CH);
    float var = 0.f;
    for (int c = 0; c < CH; ++c) { float d = src[c] - mu; var += d * d; }
    var *= (1.f / CH);
    float rstd = rsqrtf(var + 1e-5f);
    for (int c = 0; c < CH; ++c)
      hn[nrow][c] = (src[c] - mu) * rstd * g_ln[c] + b_ln[c];
  }
  __syncthreads();

  // sp = gelu(hn @ sp_in_w^T + b); image rows: cols 24..47 partial sums,
  // latent rows: cols 0..23 -> feat.
  for (int tile = wave; tile < 51; tile += 8) {   // 17 M-tiles x 3 N-tiles
    int tm = tile / 3, tn = tile % 3;
    int m = tm * 16 + fr;
    int n = tn * 16 + fr;
    v8f acc = vzero8();
    for (int k = 0; k < CH; k += 4) {
      int kk = k + koff;
      v2f a;  a.x = hn[m][kk];           a.y = hn[m][kk + 1];
      v2f bb; bb.x = w_in[n * CH + kk];  bb.y = w_in[n * CH + kk + 1];
      acc = wmma4(a, bb, acc);
    }
    if (tm < 16) {
      if (n >= 24) {  // unique writer per (tile-half, column): deterministic
        float psum = 0.f;
        for (int r = 0; r < 8; ++r) psum += geluf(acc[r] + b_in[n]);
        colpart[tm * 2 + half][n - 24] = psum;
      }
    } else {
      if (n < 24)
        for (int r = 0; r < 8; ++r)
          feat[rbase + r][n] = geluf(acc[r] + b_in[n]);
    }
  }
  __syncthreads();
  if (tid < 24) {   // fixed-order reduction over 32 partials
    float s = 0.f;
    for (int i = 0; i < 32; ++i) s += colpart[i][tid];
    colsum[tid] = s;
  }
  __syncthreads();
  for (int idx = tid; idx < LTOK * 24; idx += 256) {
    int l = idx / 24, c = idx % 24;
    feat[l][24 + c] = colsum[c] * (1.f / NIMG);
  }
  __syncthreads();

  if (tid < LTOK) {  // tiny MLP 48->24->12->1 per latent token
    int l = tid;
    float h1[24], h2[12];
    for (int o = 0; o < 24; ++o) {
      float s = b1[o];
      for (int c = 0; c < CH; ++c) s += feat[l][c] * w1[o * CH + c];
      h1[o] = geluf(s);
    }
    for (int o = 0; o < 12; ++o) {
      float s = b2[o];
      for (int c = 0; c < 24; ++c) s += h1[c] * w2[o * 24 + c];
      h2[o] = geluf(s);
    }
    float s = b3[0];
    for (int c = 0; c < 12; ++c) s += h2[c] * w3[c];
    logits[l] = s;
  }
  __syncthreads();
  if (tid == 0) {   // argmax(logits + gumbel) == argmax(p_soft)
    int best = 0;
    float bv = -1e30f;
    for (int l = 0; l < LTOK; ++l) {
      float u = gumbel_u[b * LTOK + l];
      float v = logits[l] - logf(-logf(u));
      if (v > bv) { bv = v; best = l; }
    }
    lpos = best;
  }
  __syncthreads();
  for (int n = tid; n < NTOK; n += 256)
    policy[b * NTOK + n] = (n < NIMG) ? 1.f : ((n - NIMG) < lpos ? 1.f : 0.f);
}

// ---------------------------------------------------------------------------
// 4) qkv = LN1(xs) @ qkv_w^T + b ; scatter into q (latent rows only), k, v
// ---------------------------------------------------------------------------
__global__ __launch_bounds__(256) void k_qkv(
    const float* __restrict__ xs, const float* __restrict__ g_ln,
    const float* __restrict__ b_ln, const float* __restrict__ w,
    const float* __restrict__ bias, float* __restrict__ qarr,
    float* __restrict__ karr, float* __restrict__ varr) {
  __shared__ float hn[NTOK][CH];
  int b = blockIdx.x, tid = threadIdx.x;
  int lane = tid & 31, wave = tid >> 5;
  int fr = lane & 15;
  int koff = (lane & 16) ? 2 : 0;
  int rbase = (lane & 16) ? 8 : 0;

  for (int nrow = tid; nrow < NTOK; nrow += 256) {
    const float* src = xs + ((size_t)b * NTOK + nrow) * CH;
    float mu = 0.f;
    for (int c = 0; c < CH; ++c) mu += src[c];
    mu *= (1.f / CH);
    float var = 0.f;
    for (int c = 0; c < CH; ++c) { float d = src[c] - mu; var += d * d; }
    var *= (1.f / CH);
    float rstd = rsqrtf(var + 1e-5f);
    for (int c = 0; c < CH; ++c)
      hn[nrow][c] = (src[c] - mu) * rstd * g_ln[c] + b_ln[c];
  }
  __syncthreads();

  for (int tile = wave; tile < 17 * 9; tile += 8) {  // 272x144, K=48
    int tm = tile / 9, tn = tile % 9;
    int m = tm * 16 + fr;
    int j = tn * 16 + fr;
    v8f acc = vzero8();
    for (int k = 0; k < CH; k += 4) {
      int kk = k + koff;
      v2f a;  a.x = hn[m][kk];        a.y = hn[m][kk + 1];
      v2f bb; bb.x = w[j * CH + kk];  bb.y = w[j * CH + kk + 1];
      acc = wmma4(a, bb, acc);
    }
    int s = j / CH, jj = j % CH, h = jj / HDIM, d = jj % HDIM;
    size_t bh = (size_t)b * NH + h;
    for (int r = 0; r < 8; ++r) {
      int n = tm * 16 + rbase + r;
      float val = acc[r] + bias[j];
      if (s == 0) {
        if (n >= NIMG) qarr[(bh * LTOK + (n - NIMG)) * HDIM + d] = val;
      } else if (s == 1) {
        karr[(bh * NTOK + n) * HDIM + d] = val;
      } else {
        varr[(bh * NTOK + n) * HDIM + d] = val;
      }
    }
  }
}

// ---------------------------------------------------------------------------
// 5) RoPE on k image rows (in place); cos/sin computed inline from freqs
// ---------------------------------------------------------------------------
__global__ __launch_bounds__(256) void k_rope(const float* __restrict__ freqs,
                                              float* __restrict__ karr) {
  int idx = blockIdx.x * 256 + threadIdx.x;  // b*NH*NIMG*6 = 1,572,864 exact
  int i = idx % 6;
  int n = (idx / 6) % NIMG;
  int h = (idx / (6 * NIMG)) % NH;
  int b = idx / (6 * NIMG * NH);
  float f0 = freqs[PRUNE_D * 24 + h * 6 + i];
  float f1 = freqs[192 + PRUNE_D * 24 + h * 6 + i];
  float tx = (float)(n & 15), ty = (float)(n >> 4);
  float ang = tx * f0 + ty * f1;
  float cs = cosf(ang), sn = sinf(ang);
  float* kp = karr + (((size_t)b * NH + h) * NTOK + n) * HDIM;
  float k0 = kp[2 * i], k1 = kp[2 * i + 1];
  kp[2 * i]     = k0 * cs - k1 * sn;
  kp[2 * i + 1] = k0 * sn + k1 * cs;
}

// ---------------------------------------------------------------------------
// 6) attention for the 16 latent queries: QK^T (wmma), masked softmax, P@V
//    k/v staged into LDS via CDNA5 async global->LDS loads when available.
// ---------------------------------------------------------------------------
__global__ __launch_bounds__(256) void k_attn(
    const float* __restrict__ qarr, const float* __restrict__ karr,
    const float* __restrict__ varr, const float* __restrict__ policy,
    float* __restrict__ oarr) {
  __shared__ float kT[HDIM][NTOK];
  __shared__ float vp[NTOK][16];   // v zero-padded to 16 cols
  __shared__ float S[LTOK][NTOK];
  __shared__ float pol[NTOK];
  int bh = blockIdx.x;
  int b = bh >> 2;
  int tid = threadIdx.x;
  int lane = tid & 31, wave = tid >> 5;
  int fr = lane & 15;
  int koff = (lane & 16) ? 2 : 0;
  int rbase = (lane & 16) ? 8 : 0;

  const float* kp = karr + (size_t)bh * NTOK * HDIM;
  const float* vg = varr + (size_t)bh * NTOK * HDIM;
  // async copy (transpose scatter for kT, row copy for vp)
  for (int idx = tid; idx < NTOK * HDIM; idx += 256) {
    int n = idx / HDIM, d = idx % HDIM;
    async_g2l_b32(kp + idx, &kT[d][n]);
    async_g2l_b32(vg + idx, &vp[n][d]);
  }
  for (int idx = tid; idx < NTOK * 4; idx += 256) {  // zero pad cols 12..15
    int n = idx >> 2, d = 12 + (idx & 3);
    vp[n][d] = 0.f;
  }
  for (int n = tid; n < NTOK; n += 256) pol[n] = policy[b * NTOK + n];
  async_wait_all();
  __syncthreads();

  const float* qp = qarr + (size_t)bh * LTOK * HDIM;
  const float scale = 0.28867513459481287f;  // HD^-0.5
  for (int tn = wave; tn < 17; tn += 8) {    // S = q @ k^T, K=12
    v8f acc = vzero8();
    for (int k = 0; k < HDIM; k += 4) {
      int kk = k + koff;
      v2f a;  a.x = qp[fr * HDIM + kk];     a.y = qp[fr * HDIM + kk + 1];
      v2f bb; bb.x = kT[kk][tn * 16 + fr];  bb.y = kT[kk + 1][tn * 16 + fr];
      acc = wmma4(a, bb, acc);
    }
    for (int r = 0; r < 8; ++r) S[rbase + r][tn * 16 + fr] = acc[r] * scale;
  }
  __syncthreads();

  if (tid < LTOK) {  // reference's masked-softmax variant
    int m = tid, ig = NIMG + m;
    float mx = -1e30f;
    for (int j = 0; j < NTOK; ++j) mx = fmaxf(mx, S[m][j]);  // max BEFORE mask
    float ssum = 0.f;
    for (int j = 0; j < NTOK; ++j) {
      float ap = (j == ig) ? 1.f : pol[j];
      float e = expf(S[m][j] - mx) * ap;
      S[m][j] = e + ap * 1e-6f;
      ssum += e;
    }
    float inv = 1.f / (ssum + 1e-6f);
    for (int j = 0; j < NTOK; ++j) S[m][j] *= inv;
  }
  __syncthreads();

  if (wave == 0) {   // O = P @ v, K=272 (wave-uniform branch, EXEC full)
    v8f acc = vzero8();
    for (int k = 0; k < NTOK; k += 4) {
      int kk = k + koff;
      v2f a;  a.x = S[fr][kk];   a.y = S[fr][kk + 1];
      v2f bb; bb.x = vp[kk][fr]; bb.y = vp[kk + 1][fr];
      acc = wmma4(a, bb, acc);
    }
    if (fr < HDIM) {
      for (int r = 0; r < 8; ++r) {
        int m = rbase + r;
        oarr[((size_t)b * LTOK + m) * CH + (bh & 3) * HDIM + fr] = acc[r];
      }
    }
  }
}

// ---------------------------------------------------------------------------
// 7) xsl = xs[latent rows] + o @ proj_w^T + proj_b
// ---------------------------------------------------------------------------
__global__ __launch_bounds__(128) void k_proj(
    const float* __restrict__ oarr, const float* __restrict__ w,
    const float* __restrict__ bias, const float* __restrict__ xs,
    float* __restrict__ xsl) {
  int b = blockIdx.x, tid = threadIdx.x;
  int lane = tid & 31, wave = tid >> 5;
  int fr = lane & 15;
  int koff = (lane & 16) ? 2 : 0;
  int rbase = (lane & 16) ? 8 : 0;
  const float* op = oarr + (size_t)b * LTOK * CH;
  for (int tn = wave; tn < 3; tn += 4) {  // 16x48, K=48
    int n = tn * 16 + fr;
    v8f acc = vzero8();
    for (int k = 0; k < CH; k += 4) {
      int kk = k + koff;
      v2f a;  a.x = op[fr * CH + kk]; a.y = op[fr * CH + kk + 1];
      v2f bb; bb.x = w[n * CH + kk];  bb.y = w[n * CH + kk + 1];
      acc = wmma4(a, bb, acc);
    }
    for (int r = 0; r < 8; ++r) {
      int m = rbase + r;
      xsl[((size_t)b * LTOK + m) * CH + n] =
          acc[r] + bias[n] + xs[((size_t)b * NTOK + NIMG + m) * CH + n];
    }
  }
}

// ---------------------------------------------------------------------------
// 8) MLP (LN2, fc1+gelu, fc2, residual) + final LN -> d_out (last 16 rows)
// ---------------------------------------------------------------------------
__global__ __launch_bounds__(256) void k_mlp(
    const float* __restrict__ xsl, const float* __restrict__ g2,
    const float* __restrict__ b2n, const float* __restrict__ fc1_w,
    const float* __restrict__ fc1_b, const float* __restrict__ fc2_w,
    const float* __restrict__ fc2_b, const float* __restrict__ gf,
    const float* __restrict__ bf, float* __restrict__ out) {
  __shared__ float hm[LTOK][CH];
  __shared__ float h1[LTOK][192];
  __shared__ float xo[LTOK][CH];
  int b = blockIdx.x, tid = threadIdx.x;
  int lane = tid & 31, wave = tid >> 5;
  int fr = lane & 15;
  int koff = (lane & 16) ? 2 : 0;
  int rbase = (lane & 16) ? 8 : 0;

  if (tid < LTOK) {  // LN2
    const float* src = xsl + ((size_t)b * LTOK + tid) * CH;
    float mu = 0.f;
    for (int c = 0; c < CH; ++c) mu += src[c];
    mu *= (1.f / CH);
    float var = 0.f;
    for (int c = 0; c < CH; ++c) { float d = src[c] - mu; var += d * d; }
    var *= (1.f / CH);
    float rstd = rsqrtf(var + 1e-5f);
    for (int c = 0; c < CH; ++c)
      hm[tid][c] = (src[c] - mu) * rstd * g2[c] + b2n[c];
  }
  __syncthreads();

  for (int tn = wave; tn < 12; tn += 8) {  // fc1: 16x192, K=48
    int n = tn * 16 + fr;
    v8f acc = vzero8();
    for (int k = 0; k < CH; k += 4) {
      int kk = k + koff;
      v2f a;  a.x = hm[fr][kk];           a.y = hm[fr][kk + 1];
      v2f bb; bb.x = fc1_w[n * CH + kk];  bb.y = fc1_w[n * CH + kk + 1];
      acc = wmma4(a, bb, acc);
    }
    for (int r = 0; r < 8; ++r)
      h1[rbase + r][n] = geluf(acc[r] + fc1_b[n]);
  }
  __syncthreads();

  for (int tn = wave; tn < 3; tn += 8) {  // fc2: 16x48, K=192
    int n = tn * 16 + fr;
    v8f acc = vzero8();
    for (int k = 0; k < 192; k += 4) {
      int kk = k + koff;
      v2f a;  a.x = h1[fr][kk];            a.y = h1[fr][kk + 1];
      v2f bb; bb.x = fc2_w[n * 192 + kk];  bb.y = fc2_w[n * 192 + kk + 1];
      acc = wmma4(a, bb, acc);
    }
    for (int r = 0; r < 8; ++r) {
      int m = rbase + r;
      xo[m][n] = acc[r] + fc2_b[n] + xsl[((size_t)b * LTOK + m) * CH + n];
    }
  }
  __syncthreads();

  if (tid < LTOK) {  // final LN + output
    float mu = 0.f;
    for (int c = 0; c < CH; ++c) mu += xo[tid][c];
    mu *= (1.f / CH);
    float var = 0.f;
    for (int c = 0; c < CH; ++c) { float d = xo[tid][c] - mu; var += d * d; }
    var *= (1.f / CH);
    float rstd = rsqrtf(var + 1e-5f);
    float* dst = out + ((size_t)b * LTOK + tid) * CH;
    for (int c = 0; c < CH; ++c)
      dst[c] = (xo[tid][c] - mu) * rstd * gf[c] + bf[c];
  }
}

// ---------------------------------------------------------------------------
extern "C" void kernel_launch(void* const* d_in, const int* in_sizes, int n_in,
                              void* d_out, int out_size, void* d_ws,
                              size_t ws_size, hipStream_t stream) {
  (void)in_sizes; (void)n_in; (void)out_size; (void)ws_size;
  const float* x          = (const float*)d_in[0];
  const float* conv_w     = (const float*)d_in[1];
  const float* conv_b     = (const float*)d_in[2];
  const float* pos_embed  = (const float*)d_in[3];
  const float* mask_token = (const float*)d_in[4];
  const float* lat_tok    = (const float*)d_in[5];
  const float* lat_pos    = (const float*)d_in[6];
  const float* norm_g     = (const float*)d_in[7];
  const float* norm_b     = (const float*)d_in[8];
  const float* freqs      = (const float*)d_in[9];
  const float* sp_ln_g    = (const float*)d_in[10];
  const float* sp_ln_b    = (const float*)d_in[11];
  const float* sp_in_w    = (const float*)d_in[12];
  const float* sp_in_b    = (const float*)d_in[13];
  const float* sp_w1      = (const float*)d_in[14];
  const float* sp_b1      = (const float*)d_in[15];
  const float* sp_w2      = (const float*)d_in[16];
  const float* sp_b2      = (const float*)d_in[17];
  const float* sp_w3      = (const float*)d_in[18];
  const float* sp_b3      = (const float*)d_in[19];
  const float* ln1_g      = (const float*)d_in[20];
  const float* ln1_b      = (const float*)d_in[21];
  const float* qkv_w      = (const float*)d_in[22];
  const float* qkv_b      = (const float*)d_in[23];
  const float* proj_w     = (const float*)d_in[24];
  const float* proj_b     = (const float*)d_in[25];
  const float* ln2_g      = (const float*)d_in[26];
  const float* ln2_b      = (const float*)d_in[27];
  const float* fc1_w      = (const float*)d_in[28];
  const float* fc1_b      = (const float*)d_in[29];
  const float* fc2_w      = (const float*)d_in[30];
  const float* fc2_b      = (const float*)d_in[31];
  const int*   orders     = (const int*)d_in[32];
  const float* gumbel_u   = (const float*)d_in[33];
  float* out = (float*)d_out;
  float* ws  = (float*)d_ws;

  // workspace layout (floats), total ~43 MB
  float* maskflag = ws;                                       // 65,536
  float* xs       = maskflag + (size_t)NB * NIMG;             // 3,342,336
  float* policy   = xs + (size_t)NB * NTOK * CH;              // 69,632
  float* qarr     = policy + (size_t)NB * NTOK;               // 196,608
  float* karr     = qarr + (size_t)NB * NH * LTOK * HDIM;     // 3,342,336
  float* varr     = karr + (size_t)NB * NH * NTOK * HDIM;     // 3,342,336
  float* oarr     = varr + (size_t)NB * NH * NTOK * HDIM;     // 196,608
  float* xsl      = oarr + (size_t)NB * LTOK * CH;            // 196,608

  k_mask<<<NB, 256, 0, stream>>>(orders, maskflag);
  k_embed<<<NB, 256, 0, stream>>>(x, conv_w, conv_b, pos_embed, mask_token,
                                  lat_tok, lat_pos, norm_g, norm_b, maskflag,
                                  xs);
  k_score<<<NB, 256, 0, stream>>>(xs, sp_ln_g, sp_ln_b, sp_in_w, sp_in_b,
                                  sp_w1, sp_b1, sp_w2, sp_b2, sp_w3, sp_b3,
                                  gumbel_u, policy);
  k_qkv<<<NB, 256, 0, stream>>>(xs, ln1_g, ln1_b, qkv_w, qkv_b, qarr, karr,
                                varr);
  k_rope<<<6144, 256, 0, stream>>>(freqs, karr);
  k_attn<<<NB * NH, 256, 0, stream>>>(qarr, karr, varr, policy, oarr);
  k_proj<<<NB, 128, 0, stream>>>(oarr, proj_w, proj_b, xs, xsl);
  k_mlp<<<NB, 256, 0, stream>>>(xsl, ln2_g, ln2_b, fc1_w, fc1_b, fc2_w, fc2_b,
                                norm_g, norm_b, out);
}